// LSTM_4_23433341567596
// MI455X (gfx1250) — compile-verified
//
#include <hip/hip_runtime.h>
#include <hip/hip_bf16.h>

#define SEQ_LEN 524288
#define HID 21
#define NGATE 4

typedef __attribute__((ext_vector_type(2))) float v2f;
typedef __attribute__((ext_vector_type(8))) float v8f;

__device__ __forceinline__ float fast_sigmoid(float x) {
    return 1.0f / (1.0f + __expf(-x));
}
__device__ __forceinline__ float fast_tanh(float x) {
    // tanh(x) = 1 - 2/(exp(2x)+1)
    return 1.0f - 2.0f / (1.0f + __expf(2.0f * x));
}

// Broadcast lane j's h to all lanes via v_readlane (SGPR result, no DS traffic).
__device__ __forceinline__ float lane_bcast(float v, int j) {
    return __int_as_float(__builtin_amdgcn_readlane(__float_as_int(v), j));
}

// ---------------------------------------------------------------------------
// Phase 1: the sequential LSTM recurrence. One wave32; lane k (k<21) owns
// h[k], c[k] and the four W_hh gate rows for its k (in registers). Per step:
// 21 v_readlane broadcasts + 84 scalar FMAs (v_fmac with the SGPR broadcast
// as scalar operand; 4 independent accumulator chains give the scheduler ILP
// for VOPD dual-issue). Gates are lane-aligned so the cell update needs no
// cross-lane traffic. xg is computed on the fly from a uniform 16B load of
// x_t. Lanes 21..31 store to a dump slot so the hot loop has no EXEC-mask
// branching at all.
// ---------------------------------------------------------------------------
__global__ __launch_bounds__(32) void lstm_seq_kernel(
    const float* __restrict__ src,
    const float* __restrict__ W_ih, const float* __restrict__ b_ih,
    const float* __restrict__ W_hh, const float* __restrict__ b_hh,
    float* __restrict__ hs)
{
    const int lane = threadIdx.x;
    const bool active = (lane < HID);
    const int k = active ? lane : 0;   // lanes 21..31 compute harmless garbage

    float wx[NGATE][4];
    float wh[NGATE][HID];
    float bias[NGATE];
#pragma unroll
    for (int g = 0; g < NGATE; ++g) {
        const int row = g * HID + k;
        bias[g] = b_ih[row] + b_hh[row];
#pragma unroll
        for (int j = 0; j < 4; ++j) wx[g][j] = W_ih[row * 4 + j];
#pragma unroll
        for (int j = 0; j < HID; ++j) wh[g][j] = W_hh[row * HID + j];
    }

    // lanes >= HID dump into the 32-float pad past hs[SEQ_LEN*HID]
    const int dump = SEQ_LEN * HID + lane;

    float h = 0.0f, c = 0.0f;
    const float4* __restrict__ src4 = reinterpret_cast<const float4*>(src);

    for (int t = 0; t < SEQ_LEN; ++t) {
        const float4 x = src4[t];   // uniform address -> scalar load

        float a0 = fmaf(wx[0][3], x.w, fmaf(wx[0][2], x.z, fmaf(wx[0][1], x.y, fmaf(wx[0][0], x.x, bias[0]))));
        float a1 = fmaf(wx[1][3], x.w, fmaf(wx[1][2], x.z, fmaf(wx[1][1], x.y, fmaf(wx[1][0], x.x, bias[1]))));
        float a2 = fmaf(wx[2][3], x.w, fmaf(wx[2][2], x.z, fmaf(wx[2][1], x.y, fmaf(wx[2][0], x.x, bias[2]))));
        float a3 = fmaf(wx[3][3], x.w, fmaf(wx[3][2], x.z, fmaf(wx[3][1], x.y, fmaf(wx[3][0], x.x, bias[3]))));

#pragma unroll
        for (int j = 0; j < HID; ++j) {
            const float hj = lane_bcast(h, j);   // v_readlane_b32 s, v, j
            a0 = fmaf(wh[0][j], hj, a0);
            a1 = fmaf(wh[1][j], hj, a1);
            a2 = fmaf(wh[2][j], hj, a2);
            a3 = fmaf(wh[3][j], hj, a3);
        }

        const float ig = fast_sigmoid(a0);
        const float fg = fast_sigmoid(a1);
        const float gg = fast_tanh(a2);
        const float og = fast_sigmoid(a3);
        c = fmaf(fg, c, ig * gg);
        h = og * fast_tanh(c);

        const int idx = active ? (t * HID + lane) : dump;  // v_cndmask, no EXEC churn
        hs[idx] = h;                                       // fire-and-forget store
    }
}

// ---------------------------------------------------------------------------
// Phase 2: out = hs @ W_lin.T + b_lin via V_WMMA_F32_16X16X4_F32.
// Each wave computes one 16-timestep tile: A = 16x4 chunks of hs (K padded
// 21->24, 6 WMMAs), B = W_lin^T zero-padded into 4x16, D cols 0..1 = out.
// ISA layouts: A lane L: M=L&15, K = {2*half, 2*half+1} (half = L>>4);
// D VGPR r: M = 8*half + r, N = L&15.
// Pad handling: clamp indices (always-legal address), unconditional load,
// then MULTIPLY by a 0/1 mask. A select would let the compiler sink the load
// under EXEC branches; x * mask keeps the value live -> straight-line code.
// ---------------------------------------------------------------------------
__global__ __launch_bounds__(256) void lstm_out_kernel(
    const float* __restrict__ hs,
    const float* __restrict__ W_lin,
    const float* __restrict__ b_lin,
    float* __restrict__ out)
{
    const int lane = threadIdx.x & 31;
    const int wave = threadIdx.x >> 5;
    const int tile = blockIdx.x * 8 + wave;   // 16 timesteps per tile
    const int half = lane >> 4;               // 0: K pair {0,1}, 1: K pair {2,3}
    const int n    = lane & 15;               // N index for B/D, M index for A
    const int m    = tile * 16 + n;           // timestep row this lane loads for A
    const int nrow = (n < 2) ? n : 0;         // clamped W_lin row (legal address)
    const float nmask = (n < 2) ? 1.0f : 0.0f;

    v8f acc = {};
#pragma unroll
    for (int kc = 0; kc < 6; ++kc) {
        const int kb = kc * 4 + half * 2;
        const int k0 = (kb     < HID) ? kb     : 0;   // clamped, always legal
        const int k1 = (kb + 1 < HID) ? kb + 1 : 0;
        const float m0 = (kb     < HID) ? 1.0f : 0.0f;
        const float m1 = (kb + 1 < HID) ? 1.0f : 0.0f;

        // unconditional loads (legal addresses), pad zeroed by multiply
        v2f a, b;
        a.x = hs[m * HID + k0] * m0;
        a.y = hs[m * HID + k1] * m1;
        b.x = W_lin[nrow * HID + k0] * (m0 * nmask);
        b.y = W_lin[nrow * HID + k1] * (m1 * nmask);

        acc = __builtin_amdgcn_wmma_f32_16x16x4_f32(
            /*neg_a=*/false, a, /*neg_b=*/false, b,
            /*c_mod=*/(short)0, acc, /*reuse_a=*/false, /*reuse_b=*/false);
    }

    if (n < 2) {   // only D columns 0,1 are real outputs
        const float bl = b_lin[n];
        const int rowbase = tile * 16 + half * 8;
#pragma unroll
        for (int r = 0; r < 8; ++r) {
            out[(rowbase + r) * 2 + n] = acc[r] + bl;
        }
    }
}

extern "C" void kernel_launch(void* const* d_in, const int* in_sizes, int n_in,
                              void* d_out, int out_size, void* d_ws, size_t ws_size,
                              hipStream_t stream) {
    const float* src   = (const float*)d_in[0];
    const float* W_ih  = (const float*)d_in[1];
    const float* b_ih  = (const float*)d_in[2];
    const float* W_hh  = (const float*)d_in[3];
    const float* b_hh  = (const float*)d_in[4];
    const float* W_lin = (const float*)d_in[5];
    const float* b_lin = (const float*)d_in[6];
    float* out = (float*)d_out;
    float* hs  = (float*)d_ws;   // SEQ_LEN*HID floats (+32 dump floats) < ws

    lstm_seq_kernel<<<1, 32, 0, stream>>>(src, W_ih, b_ih, W_hh, b_hh, hs);

    const int tiles  = SEQ_LEN / 16;   // 32768
    const int blocks = tiles / 8;      // 8 waves per block
    lstm_out_kernel<<<blocks, 256, 0, stream>>>(hs, W_lin, b_lin, out);
}